// GraphProp_12146167513751
// MI455X (gfx1250) — compile-verified
//
#include <hip/hip_runtime.h>

typedef float v2f __attribute__((ext_vector_type(2)));
typedef float v8f __attribute__((ext_vector_type(8)));

#define GP_ROUNDS   3
#define GP_NODE_DIM 64
#define GP_ACT_DIM  64
#define GP_EDGE_DIM 32
#define GP_N_NODES  100000
#define GP_N_EDGES  1600000
#define GP_CAT_DIM  160          // 64 (src) + 64 (dst) + 32 (edge)
#define KC_SRC  16               // K-chunks of 4 over hv[src]
#define KC_DST  16
#define KC_EDG  8
#define KC_TOT  40

// D = A(16x4) * B(4x16) + C, fp32 WMMA (wave32).
static __device__ __forceinline__ v8f wmma4(v2f a, v2f b, v8f c) {
  // 8 args: (neg_a, A, neg_b, B, c_mod, C, reuse_a, reuse_b)
  return __builtin_amdgcn_wmma_f32_16x16x4_f32(false, a, false, b, (short)0, c,
                                               false, false);
}
static __device__ __forceinline__ v2f ld2(const float* p) {
  return *(const v2f*)p;
}
static __device__ __forceinline__ float sigmoid_(float x) {
  return 1.0f / (1.0f + __expf(-x));
}

// ---------------------------------------------------------------------------
// Edge kernel: per 16-edge tile compute msg = [hv[s];hv[d];he] @ W + b and
// scatter-add into abuf[dst] (segment_sum) with global_atomic_add_f32.
// One wave per tile; 8 waves / block; N_EDGES/16 tiles exactly.
// W_msg (160x64, 40KB) is pre-swizzled into LDS so each B fragment of the
// 16x16x4 WMMA is a single ds_load_b64.
// ---------------------------------------------------------------------------
__global__ __launch_bounds__(256) void gp_edge_kernel(
    const float* __restrict__ hv, const float* __restrict__ he,
    const int* __restrict__ src, const int* __restrict__ dst,
    const float* __restrict__ W,      // 160 x 64 row-major (this round)
    const float* __restrict__ bias,   // 64
    float* __restrict__ abuf)         // N_NODES x 64 accumulator
{
  // LDS: pair index p = (c*4 + n0)*32 + lane  ->  (B.x, B.y) for that frag.
  __shared__ float Wl[KC_TOT * 4 * 32 * 2];   // 10240 floats = 40KB
  const int tid = threadIdx.x;
#pragma unroll
  for (int p = tid; p < KC_TOT * 4 * 32; p += 256) {
    const int L  = p & 31;
    const int q  = p >> 5;          // q = c*4 + n0
    const int n0 = q & 3;
    const int c  = q >> 2;
    const int k  = 4 * c + 2 * (L >> 4);
    const int col = n0 * 16 + (L & 15);
    v2f w2;
    w2.x = W[k * 64 + col];
    w2.y = W[(k + 1) * 64 + col];
    *(v2f*)&Wl[2 * p] = w2;
  }
  __syncthreads();

  const int lane = tid & 31;
  const int wave = tid >> 5;
  const int tile = blockIdx.x * 8 + wave;
  const int e0   = tile * 16;
  const int row  = lane & 15;       // A/B/C: N (and A: M) = lane % 16
  const int half = lane >> 4;       // selects K sub-pair / M upper half

  // Gather A fragments: lane holds row `row`'s features [4c+2*half, +1].
  const int sidx = src[e0 + row];
  const int didx = dst[e0 + row];
  const float* hs = hv + (long)sidx * GP_NODE_DIM + 2 * half;
  const float* hd = hv + (long)didx * GP_NODE_DIM + 2 * half;
  const float* hp = he + (long)(e0 + row) * GP_EDGE_DIM + 2 * half;

  v2f As[KC_SRC], Ad[KC_DST], Ae[KC_EDG];
#pragma unroll
  for (int c = 0; c < KC_SRC; ++c) As[c] = ld2(hs + 4 * c);
#pragma unroll
  for (int c = 0; c < KC_DST; ++c) Ad[c] = ld2(hd + 4 * c);
#pragma unroll
  for (int c = 0; c < KC_EDG; ++c) Ae[c] = ld2(hp + 4 * c);

  // dst indices for the C-layout rows this lane will scatter (M = i + 8*half)
  int drow[8];
#pragma unroll
  for (int i = 0; i < 8; ++i) drow[i] = dst[e0 + i + 8 * half];

  const v2f* Wl2 = (const v2f*)Wl;
#pragma unroll
  for (int n0 = 0; n0 < 4; ++n0) {
    const float bv = bias[n0 * 16 + row];   // bias depends on N only
    v8f acc;
#pragma unroll
    for (int i = 0; i < 8; ++i) acc[i] = bv;

    const v2f* Bp = Wl2 + (n0 * 32 + lane); // advance K-chunk: +4*32 pairs
#pragma unroll
    for (int c = 0; c < KC_SRC; ++c)
      acc = wmma4(As[c], Bp[(c) * 128], acc);
#pragma unroll
    for (int c = 0; c < KC_DST; ++c)
      acc = wmma4(Ad[c], Bp[(KC_SRC + c) * 128], acc);
#pragma unroll
    for (int c = 0; c < KC_EDG; ++c)
      acc = wmma4(Ae[c], Bp[(KC_SRC + KC_DST + c) * 128], acc);

    // segment_sum: scatter-add msg tile into abuf (L2-resident atomics)
#pragma unroll
    for (int i = 0; i < 8; ++i)
      atomicAdd(&abuf[(long)drow[i] * GP_ACT_DIM + n0 * 16 + row], acc[i]);
  }
}

// ---------------------------------------------------------------------------
// GRU kernel: per 16-node tile, gi = a @ w_ih^T + b_ih, gh = h @ w_hh^T + b_hh
// via WMMA (B fragments are contiguous float2 in row-major w_*), then gates.
// ---------------------------------------------------------------------------
__global__ __launch_bounds__(256) void gp_gru_kernel(
    const float* __restrict__ abuf, const float* __restrict__ hvin,
    const float* __restrict__ w_ih, const float* __restrict__ w_hh, // 192x64
    const float* __restrict__ b_ih, const float* __restrict__ b_hh, // 192
    float* __restrict__ hvout)
{
  const int tid  = threadIdx.x;
  const int lane = tid & 31;
  const int wave = tid >> 5;
  const int tile = blockIdx.x * 8 + wave;
  const int ntiles = GP_N_NODES / 16;
  if (tile >= ntiles) return;

  const int nbase = tile * 16;
  const int row   = lane & 15;
  const int half  = lane >> 4;

  v2f Aa[16], Ah[16];
  const float* ap = abuf + (long)(nbase + row) * GP_ACT_DIM + 2 * half;
  const float* hq = hvin + (long)(nbase + row) * GP_NODE_DIM + 2 * half;
#pragma unroll
  for (int c = 0; c < 16; ++c) { Aa[c] = ld2(ap + 4 * c); Ah[c] = ld2(hq + 4 * c); }

#pragma unroll
  for (int j0 = 0; j0 < 4; ++j0) {
    const int colb = j0 * 16 + row;           // output column within 64
    v8f ir, iz, in_, hr, hz, hn;
    {
      const float bi0 = b_ih[colb], bi1 = b_ih[64 + colb], bi2 = b_ih[128 + colb];
      const float bh0 = b_hh[colb], bh1 = b_hh[64 + colb], bh2 = b_hh[128 + colb];
#pragma unroll
      for (int i = 0; i < 8; ++i) {
        ir[i] = bi0; iz[i] = bi1; in_[i] = bi2;
        hr[i] = bh0; hz[i] = bh1; hn[i] = bh2;
      }
    }
    // B frag for column j: w[j*64 + 4c + 2*half .. +1] (contiguous)
    const float* wi0 = w_ih + (long)(colb)*64 + 2 * half;
    const float* wi1 = w_ih + (long)(64 + colb) * 64 + 2 * half;
    const float* wi2 = w_ih + (long)(128 + colb) * 64 + 2 * half;
    const float* wh0 = w_hh + (long)(colb)*64 + 2 * half;
    const float* wh1 = w_hh + (long)(64 + colb) * 64 + 2 * half;
    const float* wh2 = w_hh + (long)(128 + colb) * 64 + 2 * half;
#pragma unroll
    for (int c = 0; c < 16; ++c) {
      ir  = wmma4(Aa[c], ld2(wi0 + 4 * c), ir);
      iz  = wmma4(Aa[c], ld2(wi1 + 4 * c), iz);
      in_ = wmma4(Aa[c], ld2(wi2 + 4 * c), in_);
      hr  = wmma4(Ah[c], ld2(wh0 + 4 * c), hr);
      hz  = wmma4(Ah[c], ld2(wh1 + 4 * c), hz);
      hn  = wmma4(Ah[c], ld2(wh2 + 4 * c), hn);
    }
    // Gates. C-layout: component i -> node row M = i + 8*half, column colb.
#pragma unroll
    for (int i = 0; i < 8; ++i) {
      const long m = nbase + i + 8 * half;
      const float h = hvin[m * GP_NODE_DIM + colb];
      const float r = sigmoid_(ir[i] + hr[i]);
      const float z = sigmoid_(iz[i] + hz[i]);
      const float n = tanhf(in_[i] + r * hn[i]);
      hvout[m * GP_NODE_DIM + colb] = (1.0f - z) * n + z * h;
    }
  }
}

// ---------------------------------------------------------------------------
extern "C" void kernel_launch(void* const* d_in, const int* in_sizes, int n_in,
                              void* d_out, int out_size, void* d_ws, size_t ws_size,
                              hipStream_t stream) {
  const float* hv    = (const float*)d_in[0];
  const float* he    = (const float*)d_in[1];
  const int*   src   = (const int*)d_in[2];
  const int*   dst   = (const int*)d_in[3];
  const float* W_msg = (const float*)d_in[4];   // (3, 160, 64)
  const float* b_msg = (const float*)d_in[5];   // (3, 64)
  const float* w_ih  = (const float*)d_in[6];   // (3, 192, 64)
  const float* w_hh  = (const float*)d_in[7];   // (3, 192, 64)
  const float* b_ih  = (const float*)d_in[8];   // (3, 192)
  const float* b_hh  = (const float*)d_in[9];   // (3, 192)
  float* out = (float*)d_out;

  const size_t state_elems = (size_t)GP_N_NODES * GP_NODE_DIM;
  float* a_buf  = (float*)d_ws;                 // 25.6 MB, L2-resident
  float* hv_buf = a_buf + state_elems;          // 25.6 MB

  // hv working copy (inputs must not be mutated)
  hipMemcpyAsync(hv_buf, hv, state_elems * sizeof(float),
                 hipMemcpyDeviceToDevice, stream);

  const int edge_blocks = (GP_N_EDGES / 16) / 8;          // 12500, exact
  const int gru_blocks  = (GP_N_NODES / 16 + 7) / 8;      // 782

  for (int t = 0; t < GP_ROUNDS; ++t) {
    hipMemsetAsync(a_buf, 0, state_elems * sizeof(float), stream);
    gp_edge_kernel<<<edge_blocks, 256, 0, stream>>>(
        hv_buf, he, src, dst,
        W_msg + (size_t)t * GP_CAT_DIM * GP_ACT_DIM,
        b_msg + (size_t)t * GP_ACT_DIM, a_buf);
    float* outp = (t == GP_ROUNDS - 1) ? out : hv_buf;
    gp_gru_kernel<<<gru_blocks, 256, 0, stream>>>(
        a_buf, hv_buf,
        w_ih + (size_t)t * 3 * GP_NODE_DIM * GP_NODE_DIM,
        w_hh + (size_t)t * 3 * GP_NODE_DIM * GP_NODE_DIM,
        b_ih + (size_t)t * 3 * GP_NODE_DIM,
        b_hh + (size_t)t * 3 * GP_NODE_DIM, outp);
  }
}